// SSD4PointLoss_36043365548644
// MI455X (gfx1250) — compile-verified
//
#include <hip/hip_runtime.h>
#include <math.h>

// ---------------- problem constants ----------------
constexpr int Bn = 8;
constexpr int An = 98304;
constexpr int Cn = 81;
constexpr int Gn = 64;
constexpr float IOU_T = 0.45f;
constexpr int NEGPOS = 3;

// ---------------- workspace layout (bytes) ----------------
constexpr size_t OFF_AABB    = 0;
constexpr size_t SZ_AABB     = (size_t)Bn * Gn * 4 * 4;      // 8 KB
constexpr size_t OFF_MATCHED = OFF_AABB + SZ_AABB;
constexpr size_t SZ_MATCHED  = (size_t)Bn * An * 4;          // 3 MB
constexpr size_t OFF_NEGV    = OFF_MATCHED + SZ_MATCHED;
constexpr size_t SZ_NEGV     = (size_t)Bn * An * 4;          // 3 MB
constexpr size_t OFF_PACK    = OFF_NEGV + SZ_NEGV;           // 8B aligned
constexpr size_t SZ_PACK     = (size_t)Bn * Gn * 8;
constexpr size_t OFF_ACC     = OFF_PACK + SZ_PACK;           // 5*Bn floats
constexpr size_t SZ_ACC      = (size_t)5 * Bn * 4;
constexpr size_t OFF_STATE   = OFF_ACC + SZ_ACC;             // Bn * 4 u32
constexpr size_t SZ_STATE    = (size_t)Bn * 4 * 4;
constexpr size_t OFF_H0      = OFF_STATE + SZ_STATE;
constexpr size_t SZ_H0       = (size_t)Bn * 4096 * 4;
constexpr size_t OFF_H1      = OFF_H0 + SZ_H0;
constexpr size_t SZ_H1       = SZ_H0;
constexpr size_t OFF_H2      = OFF_H1 + SZ_H1;
constexpr size_t SZ_H2       = (size_t)Bn * 256 * 4;
constexpr size_t WS_END      = OFF_H2 + SZ_H2;
constexpr size_t ZERO_OFF    = OFF_PACK;
constexpr int    ZERO_WORDS  = (int)((WS_END - OFF_PACK) / 4);

#define DEVI __device__ __forceinline__

typedef float v2f __attribute__((ext_vector_type(2)));
typedef float v8f __attribute__((ext_vector_type(8)));

// ---------------- wave helpers (wave32) ----------------
DEVI float wave_fmax(float v) {
#pragma unroll
  for (int o = 16; o > 0; o >>= 1) v = fmaxf(v, __shfl_xor(v, o, 32));
  return v;
}
DEVI float wave_fsum(float v) {
#pragma unroll
  for (int o = 16; o > 0; o >>= 1) v += __shfl_xor(v, o, 32);
  return v;
}
DEVI unsigned long long shflx_u64(unsigned long long v, int m) {
  unsigned lo = (unsigned)(v & 0xFFFFFFFFull);
  unsigned hi = (unsigned)(v >> 32);
  lo = (unsigned)__shfl_xor((int)lo, m, 32);
  hi = (unsigned)__shfl_xor((int)hi, m, 32);
  return ((unsigned long long)hi << 32) | lo;
}

// Wave-wide f32 sum via V_WMMA_F32_16X16X4_F32:
// A(16x4): lane m in [0,15] puts v in K=0 slot (VGPR0), lanes 16-31 in K=2 slot.
// B = ones(4x16) => D[m][n] = v[m] + v[m+16] for every n.
// Sum the 8 D registers per lane (rows m..m+7), then add across half-waves.
DEVI float wmma_wave_sum(float v) {
  v2f a; a[0] = v;   a[1] = 0.0f;
  v2f b; b[0] = 1.0f; b[1] = 1.0f;
  v8f c = {0.f, 0.f, 0.f, 0.f, 0.f, 0.f, 0.f, 0.f};
  c = __builtin_amdgcn_wmma_f32_16x16x4_f32(false, a, false, b, (short)0, c,
                                            false, false);
  float s = c[0] + c[1] + c[2] + c[3] + c[4] + c[5] + c[6] + c[7];
  s += __shfl_xor(s, 16, 32);
  return s;
}

DEVI float smoothl1(float d) {
  float ad = fabsf(d);
  return ad < 1.f ? 0.5f * d * d : ad - 0.5f;
}

// ---------------- kernels ----------------
__global__ void k_init(unsigned* z, int n) {
  int stride = gridDim.x * blockDim.x;
  for (int i = blockIdx.x * blockDim.x + threadIdx.x; i < n; i += stride)
    z[i] = 0u;
}

__global__ void k_aabb(const float* __restrict__ gt, float* __restrict__ aabb) {
  int i = blockIdx.x * blockDim.x + threadIdx.x;
  if (i >= Bn * Gn) return;
  const float* p = gt + (size_t)i * 8;
  float xmn = fminf(fminf(p[0], p[2]), fminf(p[4], p[6]));
  float xmx = fmaxf(fmaxf(p[0], p[2]), fmaxf(p[4], p[6]));
  float ymn = fminf(fminf(p[1], p[3]), fminf(p[5], p[7]));
  float ymx = fmaxf(fmaxf(p[1], p[3]), fmaxf(p[5], p[7]));
  float* o = aabb + (size_t)i * 4;
  o[0] = xmn; o[1] = ymn; o[2] = xmx; o[3] = ymx;
}

// Per-anchor best GT (argmax over g, first-index tie) and per-GT best anchor
// (packed u64 atomicMax over all anchors; ~a in low bits -> smallest index wins ties).
constexpr int APT = 4;       // anchors per thread
constexpr int MBLK = 256;    // threads per block
__global__ void k_match(const float* __restrict__ anchors,
                        const float* __restrict__ aabb,
                        int* __restrict__ matched,
                        unsigned long long* __restrict__ pack) {
  int b = blockIdx.y;
  int t = threadIdx.x;
  __shared__ float sg[Gn * 4];
  __shared__ unsigned long long swp[MBLK / 32];
  sg[t] = aabb[(size_t)b * Gn * 4 + t];   // Gn*4 == 256 == blockDim
  __syncthreads();

  int base = blockIdx.x * (MBLK * APT);
  const float4* an4 = (const float4*)(anchors + (size_t)b * An * 4);
  float4 an[APT];
  int aidx[APT];
  float aa[APT], bestv[APT];
  int bestg[APT];
#pragma unroll
  for (int i = 0; i < APT; ++i) {
    aidx[i] = base + i * MBLK + t;
    an[i] = an4[aidx[i]];
    aa[i] = (an[i].z - an[i].x) * (an[i].w - an[i].y);
    bestv[i] = -1.0f;
    bestg[i] = 0;
  }
  int wave = t >> 5, lane = t & 31;

  for (int g = 0; g < Gn; ++g) {
    float gx1 = sg[g * 4 + 0], gy1 = sg[g * 4 + 1];
    float gx2 = sg[g * 4 + 2], gy2 = sg[g * 4 + 3];
    float ag = (gx2 - gx1) * (gy2 - gy1);
    unsigned long long lp = 0ull;
#pragma unroll
    for (int i = 0; i < APT; ++i) {
      float lx = fmaxf(gx1, an[i].x), ly = fmaxf(gy1, an[i].y);
      float rx = fminf(gx2, an[i].z), ry = fminf(gy2, an[i].w);
      float w = fmaxf(rx - lx, 0.f), h = fmaxf(ry - ly, 0.f);
      float inter = w * h;
      float iou = inter / (ag + aa[i] - inter);
      if (iou > bestv[i]) { bestv[i] = iou; bestg[i] = g; }
      unsigned long long p =
          ((unsigned long long)__float_as_uint(iou) << 32) |
          (unsigned)(~aidx[i]);
      lp = (p > lp) ? p : lp;
    }
#pragma unroll
    for (int o = 16; o > 0; o >>= 1) {
      unsigned long long q = shflx_u64(lp, o);
      lp = (q > lp) ? q : lp;
    }
    if (lane == 0) swp[wave] = lp;
    __syncthreads();
    if (t == 0) {
      unsigned long long m = swp[0];
#pragma unroll
      for (int w2 = 1; w2 < MBLK / 32; ++w2) m = (swp[w2] > m) ? swp[w2] : m;
      atomicMax(&pack[b * Gn + g], m);
    }
    __syncthreads();
  }
#pragma unroll
  for (int i = 0; i < APT; ++i)
    matched[(size_t)b * An + aidx[i]] = (bestv[i] < IOU_T) ? -1 : bestg[i];
}

// Force-match: matched[best_anchor[g]] = g, sequential per batch (last-wins).
__global__ void k_force(const unsigned long long* __restrict__ pack,
                        int* __restrict__ matched) {
  int b = blockIdx.x * blockDim.x + threadIdx.x;
  if (b >= Bn) return;
  for (int g = 0; g < Gn; ++g) {
    unsigned a = ~((unsigned)(pack[b * Gn + g] & 0xFFFFFFFFull));
    matched[(size_t)b * An + a] = g;
  }
}

// One wave per anchor: coalesced streaming of the 255 MB logits (BW-bound).
__global__ void k_cls(const float* __restrict__ logits,
                      const int* __restrict__ matched,
                      const int* __restrict__ labels,
                      float* __restrict__ negv,
                      float* __restrict__ clsfg_sum) {
  int b = blockIdx.y;
  int wave = threadIdx.x >> 5, lane = threadIdx.x & 31;
  int wpg = gridDim.x * (blockDim.x >> 5);
  int wgid = blockIdx.x * (blockDim.x >> 5) + wave;
  float accfg = 0.f;
  for (int a = wgid; a < An; a += wpg) {
    const float* row = logits + ((size_t)b * An + a) * Cn;
    __builtin_prefetch(row + (size_t)wpg * Cn, 0, 0);
    float x0 = row[lane];
    float x1 = row[lane + 32];
    bool has2 = (lane + 64) < Cn;
    float x2 = has2 ? row[lane + 64] : -3.0e38f;
    float m = wave_fmax(fmaxf(fmaxf(x0, x1), x2));
    float s = __expf(x0 - m) + __expf(x1 - m) + (has2 ? __expf(x2 - m) : 0.f);
    s = wave_fsum(s);
    if (lane == 0) {
      float lse = m + __logf(s);
      int mm = matched[(size_t)b * An + a];
      bool fg = mm >= 0;
      int mi = fg ? mm : 0;
      int t = fg ? labels[b * Gn + mi] : (Cn - 1);
      float cl = lse - row[t];
      negv[(size_t)b * An + a] = fg ? -1.0f : cl;
      if (fg) accfg += cl;
    }
  }
  __shared__ float sacc[8];
  if (lane == 0) sacc[wave] = accfg;
  __syncthreads();
  if (threadIdx.x == 0) {
    float s = 0.f;
    for (int i = 0; i < (int)(blockDim.x >> 5); ++i) s += sacc[i];
    atomicAdd(&clsfg_sum[b], s);
  }
}

// Smooth-L1 box loss on foreground anchors + fg count; WMMA reduction.
__global__ void k_bbox(const float* __restrict__ anchors,
                       const float* __restrict__ gt,
                       const float* __restrict__ reg,
                       const int* __restrict__ matched,
                       float* __restrict__ bbox_sum,
                       float* __restrict__ fgcnt) {
  int b = blockIdx.y;
  int tid = blockIdx.x * blockDim.x + threadIdx.x;
  int stride = gridDim.x * blockDim.x;
  const float4* an4 = (const float4*)(anchors + (size_t)b * An * 4);
  float acc = 0.f, cnt = 0.f;
  for (int a = tid; a < An; a += stride) {
    int mm = matched[(size_t)b * An + a];
    if (mm >= 0) {
      cnt += 1.f;
      float4 an = an4[a];
      const float* g8 = gt + ((size_t)b * Gn + mm) * 8;
      const float* r8 = reg + ((size_t)b * An + a) * 8;
      float acx = (an.x + an.z) * 0.5f, acy = (an.y + an.w) * 0.5f;
      float aw = an.z - an.x, ah = an.w - an.y;
      float s = 0.f;
#pragma unroll
      for (int k = 0; k < 4; ++k) {
        float tx = (g8[2 * k] - acx) / aw;
        float ty = (g8[2 * k + 1] - acy) / ah;
        s += smoothl1(r8[2 * k] - tx) + smoothl1(r8[2 * k + 1] - ty);
      }
      acc += s;
    }
  }
  float wacc = wmma_wave_sum(acc);
  float wcnt = wmma_wave_sum(cnt);
  __shared__ float s1[8], s2[8];
  int wave = threadIdx.x >> 5;
  if ((threadIdx.x & 31) == 0) { s1[wave] = wacc; s2[wave] = wcnt; }
  __syncthreads();
  if (threadIdx.x == 0) {
    float A = 0.f, C2 = 0.f;
    for (int i = 0; i < 8; ++i) { A += s1[i]; C2 += s2[i]; }
    atomicAdd(&bbox_sum[b], A);
    atomicAdd(&fgcnt[b], C2);
  }
}

// Exact 3-level radix-select over float bits (values >= 0 -> monotone bits).
template <int LEVEL>
__global__ void k_hist(const float* __restrict__ negv,
                       const unsigned* __restrict__ state,
                       unsigned* __restrict__ hist) {
  constexpr int NB = (LEVEL == 2) ? 256 : 4096;
  int b = blockIdx.y;
  __shared__ unsigned h[NB];
  for (int i = threadIdx.x; i < NB; i += blockDim.x) h[i] = 0u;
  __syncthreads();
  unsigned pre = state[b * 4 + 1];
  int tid = blockIdx.x * blockDim.x + threadIdx.x;
  int stride = gridDim.x * blockDim.x;
  for (int a = tid; a < An; a += stride) {
    float v = negv[(size_t)b * An + a];
    if (v < 0.f) continue;   // fg sentinel (-1) excluded
    unsigned bits = __float_as_uint(v);
    if (LEVEL == 0) {
      atomicAdd(&h[bits >> 20], 1u);
    } else if (LEVEL == 1) {
      if ((bits >> 20) == pre) atomicAdd(&h[(bits >> 8) & 0xFFFu], 1u);
    } else {
      if ((bits >> 8) == pre) atomicAdd(&h[bits & 0xFFu], 1u);
    }
  }
  __syncthreads();
  for (int i = threadIdx.x; i < NB; i += blockDim.x) {
    unsigned c = h[i];
    if (c) atomicAdd(&hist[(size_t)b * NB + i], c);
  }
}

template <int LEVEL>
__global__ void k_scan(const float* __restrict__ fgcnt,
                       unsigned* __restrict__ state,
                       const unsigned* __restrict__ hist,
                       float* __restrict__ tie_add) {
  int b = blockIdx.x * blockDim.x + threadIdx.x;
  if (b >= Bn) return;
  constexpr int NB = (LEVEL == 2) ? 256 : 4096;
  int Krem = (LEVEL == 0) ? NEGPOS * (int)(fgcnt[b] + 0.5f)
                          : (int)state[b * 4 + 0];
  unsigned prefix = (LEVEL == 0) ? 0u : state[b * 4 + 1];
  const unsigned* h = hist + (size_t)b * NB;
  if (Krem <= 0) {
    state[b * 4 + 0] = 0u;
    if (LEVEL < 2) state[b * 4 + 1] = prefix << 12;
    else { state[b * 4 + 2] = 0xFFFFFFFFu; tie_add[b] = 0.f; }
    return;
  }
  long long acc = 0;
  int chosen = 0, kn = Krem;
  bool found = false;
  for (int i = NB - 1; i >= 0; --i) {
    unsigned c = h[i];
    if (acc + (long long)c >= (long long)Krem) {
      chosen = i; kn = Krem - (int)acc; found = true; break;
    }
    acc += c;
  }
  if (!found) { chosen = 0; kn = Krem - (int)acc + (int)h[0]; }  // K > #neg
  if (LEVEL < 2) {
    state[b * 4 + 0] = (unsigned)kn;
    state[b * 4 + 1] = (prefix << 12) | (unsigned)chosen;
  } else {
    unsigned T = (prefix << 8) | (unsigned)chosen;
    state[b * 4 + 2] = T;
    int cat = (int)h[chosen];
    int tie = kn < cat ? kn : cat;
    if (tie < 0) tie = 0;
    tie_add[b] = (float)tie * __uint_as_float(T);
  }
}

__global__ void k_negsum(const float* __restrict__ negv,
                         const unsigned* __restrict__ state,
                         float* __restrict__ neg_sum) {
  int b = blockIdx.y;
  unsigned T = state[b * 4 + 2];
  int tid = blockIdx.x * blockDim.x + threadIdx.x;
  int stride = gridDim.x * blockDim.x;
  float acc = 0.f;
  for (int a = tid; a < An; a += stride) {
    float v = negv[(size_t)b * An + a];
    if (v >= 0.f && __float_as_uint(v) > T) acc += v;
  }
  float w = wmma_wave_sum(acc);
  __shared__ float s1[8];
  int wave = threadIdx.x >> 5;
  if ((threadIdx.x & 31) == 0) s1[wave] = w;
  __syncthreads();
  if (threadIdx.x == 0) {
    float s = 0.f;
    for (int i = 0; i < 8; ++i) s += s1[i];
    atomicAdd(&neg_sum[b], s);
  }
}

__global__ void k_final(const float* __restrict__ acc, float* __restrict__ out) {
  if (threadIdx.x != 0 || blockIdx.x != 0) return;
  float fgt = 0.f, bb = 0.f, cf = 0.f, ns = 0.f;
  for (int b = 0; b < Bn; ++b) {
    bb += acc[b];
    cf += acc[8 + b];
    fgt += acc[16 + b];
    ns += acc[24 + b] + acc[32 + b];
  }
  float N = fmaxf(1.f, fgt);
  float bt = bb / N;
  float ct = (cf + ns) / N;
  out[0] = bt + ct;
  out[1] = bt;
  out[2] = ct;
}

// ---------------- host launch ----------------
extern "C" void kernel_launch(void* const* d_in, const int* in_sizes, int n_in,
                              void* d_out, int out_size, void* d_ws,
                              size_t ws_size, hipStream_t stream) {
  const float* cls_logits = (const float*)d_in[0];
  const float* bbox_reg   = (const float*)d_in[1];
  const float* anchors    = (const float*)d_in[2];
  const float* gt_boxes   = (const float*)d_in[3];
  const int*   gt_labels  = (const int*)d_in[4];
  float* out = (float*)d_out;
  char* ws = (char*)d_ws;

  float* aabb = (float*)(ws + OFF_AABB);
  int* matched = (int*)(ws + OFF_MATCHED);
  float* negv = (float*)(ws + OFF_NEGV);
  unsigned long long* pack = (unsigned long long*)(ws + OFF_PACK);
  float* acc = (float*)(ws + OFF_ACC);     // [bbox|clsfg|fgcnt|negsum|tie] x Bn
  unsigned* state = (unsigned*)(ws + OFF_STATE);
  unsigned* h0 = (unsigned*)(ws + OFF_H0);
  unsigned* h1 = (unsigned*)(ws + OFF_H1);
  unsigned* h2 = (unsigned*)(ws + OFF_H2);

  k_init<<<256, 256, 0, stream>>>((unsigned*)(ws + ZERO_OFF), ZERO_WORDS);
  k_aabb<<<(Bn * Gn + 255) / 256, 256, 0, stream>>>(gt_boxes, aabb);
  k_match<<<dim3(An / (MBLK * APT), Bn), MBLK, 0, stream>>>(anchors, aabb,
                                                            matched, pack);
  k_force<<<1, 32, 0, stream>>>(pack, matched);
  k_cls<<<dim3(512, Bn), 256, 0, stream>>>(cls_logits, matched, gt_labels,
                                           negv, acc + 8);
  k_bbox<<<dim3(192, Bn), 256, 0, stream>>>(anchors, gt_boxes, bbox_reg,
                                            matched, acc + 0, acc + 16);
  k_hist<0><<<dim3(96, Bn), 256, 0, stream>>>(negv, state, h0);
  k_scan<0><<<1, 32, 0, stream>>>(acc + 16, state, h0, acc + 32);
  k_hist<1><<<dim3(96, Bn), 256, 0, stream>>>(negv, state, h1);
  k_scan<1><<<1, 32, 0, stream>>>(acc + 16, state, h1, acc + 32);
  k_hist<2><<<dim3(96, Bn), 256, 0, stream>>>(negv, state, h2);
  k_scan<2><<<1, 32, 0, stream>>>(acc + 16, state, h2, acc + 32);
  k_negsum<<<dim3(96, Bn), 256, 0, stream>>>(negv, state, acc + 24);
  k_final<<<1, 1, 0, stream>>>(acc, out);
}